// HyperRadialNeuralFourierCelularAutomata_77300821393978
// MI455X (gfx1250) — compile-verified
//
#include <hip/hip_runtime.h>
#include <cstdint>

// ---- problem constants (fixed by the reference) ----
#define NBITS      32
#define HDC        512
#define DD         529           // 23*23
#define BATCH      2
#define BD         1058          // BATCH*DD
#define ROWS_PER_B 16928         // DD*NBITS
#define TOTROWS    33856         // BATCH*DD*NBITS
#define PROJROWS   529           // rows with possibly-nonzero proj (270848/512)
#define MASKWORDS  8464          // 270848 bits / 32
#define BIN        17334272      // per-channel bin elements = TOTROWS*HDC
#define HDR        5290          // 5*BD header floats

typedef float v4f __attribute__((ext_vector_type(4)));

// Kernel 1: zero proj bitmask, compute r/g/b/a/s header floats, and pack the
// bitcast words (one per (b,d) per channel) into workspace for the fill kernels.
__global__ void k_init(const float* __restrict__ data,
                       const float* __restrict__ structure,
                       const float* __restrict__ params,
                       float* __restrict__ out,
                       uint32_t* __restrict__ ws) {
  int i = blockIdx.x * 256 + threadIdx.x;
  if (i < MASKWORDS) ws[i] = 0u;
  if (i < BD) {
    int b = i / DD;
    int d = i - b * DD;
    const float* base = data + b * 4 * DD;
    float r  = base[d];
    float g  = base[DD + d];
    float bl = base[2 * DD + d];
    float a  = base[3 * DD + d];
    float s  = structure[i] * params[d];
    out[i]           = r;
    out[BD + i]      = g;
    out[2 * BD + i]  = bl;
    out[3 * BD + i]  = a;
    out[4 * BD + i]  = s;
    uint32_t* wb = ws + MASKWORDS;
    wb[0 * BD + i] = __float_as_uint(r);
    wb[1 * BD + i] = __float_as_uint(g);
    wb[2 * BD + i] = __float_as_uint(bl);
    wb[3 * BD + i] = __float_as_uint(a);
    wb[4 * BD + i] = __float_as_uint(s);
  }
}

// Kernel 2: scatter non_zero_indices into the packed bitmask (OR is commutative
// -> deterministic regardless of atomic ordering / duplicate indices).
__global__ void k_scatter(const int* __restrict__ nz, int n,
                          uint32_t* __restrict__ mask) {
  int i = blockIdx.x * 256 + threadIdx.x;
  if (i < n) {
    uint32_t idx = (uint32_t)nz[i];
    atomicOr((unsigned int*)&mask[idx >> 5], 1u << (idx & 31u));
  }
}

// Kernel 3a: the 529 rows where proj can be nonzero (all in b=0).
// Stage the row's 64-byte bitmask slice into LDS via the CDNA5 async
// global->LDS path, then emit bit^proj as B128 stores.
__global__ void k_fill_proj(const uint32_t* __restrict__ ws,
                            float* __restrict__ out) {
  __shared__ __align__(16) uint32_t sm[16];
  int row = blockIdx.x;        // 0..528  (b = 0)
  int t   = threadIdx.x;       // 0..127

  if (t < 4) {
    uint64_t ga = (uint64_t)(uintptr_t)(ws + row * 16 + t * 4);
    uint32_t la = (uint32_t)(uintptr_t)(&sm[t * 4]);   // LDS byte offset
    asm volatile("global_load_async_to_lds_b128 %0, %1, off"
                 :: "v"(la), "v"(ga) : "memory");
  }
  asm volatile("s_wait_asynccnt 0x0" ::: "memory");
  __syncthreads();

  int d   = row >> 5;
  int bit = row & 31;
  const uint32_t* wb = ws + MASKWORDS;
  int h0 = t * 4;
  uint32_t pm = (sm[h0 >> 5] >> (h0 & 31)) & 0xFu;   // 4 proj bits
  float p0 = (float)(pm & 1u);
  float p1 = (float)((pm >> 1) & 1u);
  float p2 = (float)((pm >> 2) & 1u);
  float p3 = (float)((pm >> 3) & 1u);

  float* o = out + HDR + (size_t)row * HDC + h0;
#pragma unroll
  for (int ch = 0; ch < 5; ++ch) {
    uint32_t bv = (wb[ch * BD + d] >> bit) & 1u;     // b=0 -> packed idx = d
    float fb = (float)bv;
    v4f v;
    v.x = (fb == p0) ? 0.0f : 1.0f;
    v.y = (fb == p1) ? 0.0f : 1.0f;
    v.z = (fb == p2) ? 0.0f : 1.0f;
    v.w = (fb == p3) ? 0.0f : 1.0f;
    *(v4f*)(o + (size_t)ch * BIN) = v;
  }
}

// Kernel 3b: remaining 33,327 rows: proj == 0, output is a per-channel constant
// broadcast -> pure non-temporal B128 store stream (store-bandwidth bound).
__global__ void k_fill_rest(const uint32_t* __restrict__ ws,
                            float* __restrict__ out) {
  int t    = threadIdx.x;                 // 0..255, 2 rows per block
  int row  = PROJROWS + blockIdx.x * 2 + (t >> 7);
  if (row >= TOTROWS) return;
  int lane = t & 127;

  int b   = row / ROWS_PER_B;
  int rr  = row - b * ROWS_PER_B;
  int d   = rr >> 5;
  int bit = rr & 31;
  int idx = b * DD + d;

  const uint32_t* wb = ws + MASKWORDS;
  int h0 = lane * 4;
  float* o = out + HDR + (size_t)row * HDC + h0;
#pragma unroll
  for (int ch = 0; ch < 5; ++ch) {
    float f = (float)((wb[ch * BD + idx] >> bit) & 1u);
    v4f v = {f, f, f, f};
    __builtin_nontemporal_store(v, (v4f*)(o + (size_t)ch * BIN));
  }
}

extern "C" void kernel_launch(void* const* d_in, const int* in_sizes, int n_in,
                              void* d_out, int out_size, void* d_ws, size_t ws_size,
                              hipStream_t stream) {
  const float* data      = (const float*)d_in[0];
  const float* structure = (const float*)d_in[1];
  const float* params    = (const float*)d_in[2];
  const int*   nz        = (const int*)d_in[3];
  int nnz = in_sizes[3];

  float*    out = (float*)d_out;
  uint32_t* ws  = (uint32_t*)d_ws;

  // 1) zero bitmask + header + packed channel words
  k_init<<<(MASKWORDS + 255) / 256, 256, 0, stream>>>(data, structure, params, out, ws);
  // 2) sparse scatter into bitmask
  k_scatter<<<(nnz + 255) / 256, 256, 0, stream>>>(nz, nnz, ws);
  // 3a) proj-region rows (async LDS staging of bitmask)
  k_fill_proj<<<PROJROWS, 128, 0, stream>>>(ws, out);
  // 3b) broadcast rows (NT B128 store stream)
  k_fill_rest<<<(TOTROWS - PROJROWS + 1) / 2, 256, 0, stream>>>(ws, out);
}